// LLamaAttention_29592324669789
// MI455X (gfx1250) — compile-verified
//
#include <hip/hip_runtime.h>

typedef __attribute__((ext_vector_type(2))) float v2f;
typedef __attribute__((ext_vector_type(8))) float v8f;
typedef int v4i_ __attribute__((vector_size(4 * sizeof(int))));

#define B_ 64
#define H_ 32
#define HKV_ 4
#define D_ 128
#define G_ 8
#define MAX_BLOCKS_ 128
#define NCHUNK_ 8
#define CHUNK_TOK_ 256
#define KVROW_ (HKV_ * D_)
#define LROW_ 132  /* padded LDS row stride (floats): bank-conflict-free frags */
#define SCALE_ 0.088388347648318447f
#define NEGINF_ -1.0e30f

#define HAS_ASYNC_LDS __has_builtin(__builtin_amdgcn_global_load_async_to_lds_b128)

// global (device, AS1) source and LDS (AS3) destination for async b128 copy
#define AS1V(p) ((__attribute__((address_space(1))) v4i_*)(p))
#define AS3V(p) ((__attribute__((address_space(3))) v4i_*)(p))

// One workgroup = one (batch, kv-head). 8 waves = 8 sequence chunks of 256
// tokens (flash-decoding). Each wave does online-softmax attention over its
// chunk with f32 WMMA for both GEMMs; K/V tiles are staged to LDS with
// gfx1250 async b128 copies; waves merge via LDS log-sum-exp at the end.
__global__ __launch_bounds__(256) void paged_attn_decode(
    const float* __restrict__ Q, const float* __restrict__ K,
    const float* __restrict__ V, const float* __restrict__ Kcache,
    const float* __restrict__ Vcache, const float* __restrict__ cosb,
    const float* __restrict__ sinb, const int* __restrict__ input_length,
    const int* __restrict__ fetch_slots, float* __restrict__ out) {
  // Stage area (main loop) and combine area (tail) share the same LDS.
  union SMem {
    float stage[NCHUNK_][2][16][LROW_];  // [wave][K/V][row][dim(+pad)]
    float red[NCHUNK_][32][66];          // [wave][lane][64 acc | m | l]
  };
  __shared__ SMem sm;

  const int bid = blockIdx.x;
  const int b = bid / HKV_, kvh = bid % HKV_;
  const int tid = threadIdx.x;
  const int wave = tid >> 5, lane = tid & 31;
  const int h = lane & 15;   // head column this lane owns (valid if < 8)
  const int sel = lane >> 4; // which half-tile of tokens this lane owns

  const int N = input_length[b];
  const int Ncache = N - 1;  // tokens served from cache; N-1 is spliced
  const int t0 = wave * CHUNK_TOK_;
  int t1 = t0 + CHUNK_TOK_;
  if (t1 > Ncache) t1 = Ncache;
  const int* fs = fetch_slots + b * MAX_BLOCKS_;
  const float* cb = cosb + (size_t)b * D_;
  const float* sb = sinb + (size_t)b * D_;

  // ---- RoPE'd, pre-scaled Q fragments in WMMA B-operand layout ----
  // B 4x16 layout per K=4 slice c: v0 = B[4c + 2*sel][head=lane%16],
  //                                v1 = B[4c + 2*sel + 1][head]
  float2 qf[32];
  {
    const int hg = h * HKV_ + kvh;  // query head for this kv head & group slot
    const float* qp = Q + ((size_t)b * H_ + hg) * D_;
#pragma unroll
    for (int c = 0; c < 32; ++c) {
      const int d0 = 4 * c + 2 * sel;
      float fx = 0.f, fy = 0.f;
      if (h < G_) {
        const float x0 = qp[d0], x1 = qp[d0 + 1];
        const float r0 = (d0 < 64) ? -qp[d0 + 64] : qp[d0 - 64];
        const float r1 = (d0 < 64) ? -qp[d0 + 65] : qp[d0 - 63];
        fx = SCALE_ * (x0 * cb[d0] + r0 * sb[d0]);
        fy = SCALE_ * (x1 * cb[d0 + 1] + r1 * sb[d0 + 1]);
      }
      qf[c].x = fx;
      qf[c].y = fy;
    }
  }

  // O^T accumulators: oacc[cc] holds dims [16cc+8*sel .. +8) for head h
  v8f oacc[8];
#pragma unroll
  for (int cc = 0; cc < 8; ++cc)
    oacc[cc] = v8f{0.f, 0.f, 0.f, 0.f, 0.f, 0.f, 0.f, 0.f};
  float m_run = NEGINF_, l_run = 0.f;

  const int nt = (t1 > t0) ? ((t1 - t0 + 15) >> 4) : 0;
  for (int i = 0; i < nt; ++i) {
    const int slot = fs[(t0 >> 4) + i];  // 16-token tile == one cache block
    if (i + 1 < nt) {
      const int sn = fs[(t0 >> 4) + i + 1];
      __builtin_prefetch(Kcache + (size_t)(sn + h) * KVROW_ + kvh * D_ + 2 * sel, 0, 3);
      __builtin_prefetch(Vcache + (size_t)(sn + h) * KVROW_ + kvh * D_ + 2 * sel, 0, 3);
    }

#if HAS_ASYNC_LDS
    // ---- stage K and V tiles to LDS: one dense 512B row per async b128 ----
    {
      const float* kg = Kcache + (size_t)slot * KVROW_ + kvh * D_ + lane * 4;
      const float* vg = Vcache + (size_t)slot * KVROW_ + kvh * D_ + lane * 4;
      // prior tile's ds reads must retire before the async engine rewrites LDS
      asm volatile("s_wait_dscnt 0x0" ::: "memory");
#pragma unroll
      for (int r = 0; r < 16; ++r) {
        __builtin_amdgcn_global_load_async_to_lds_b128(
            AS1V(kg + (size_t)r * KVROW_),
            AS3V(&sm.stage[wave][0][r][lane * 4]), 0, 0);
        __builtin_amdgcn_global_load_async_to_lds_b128(
            AS1V(vg + (size_t)r * KVROW_),
            AS3V(&sm.stage[wave][1][r][lane * 4]), 0, 0);
      }
#if __has_builtin(__builtin_amdgcn_s_wait_asynccnt)
      __builtin_amdgcn_s_wait_asynccnt(0);
#else
      asm volatile("s_wait_asynccnt 0x0" ::: "memory");
#endif
    }
    const float* kfr = &sm.stage[wave][0][h][0];  // K row h (padded stride)
#else
    const float* kfr = Kcache + (size_t)(slot + h) * KVROW_ + kvh * D_;
    const float* vbg = Vcache + (size_t)slot * KVROW_ + kvh * D_;
#endif

    // ---- S^T = K_tile(16x128) x Q^T(128x16heads), chained K=4 WMMAs ----
    v8f s = v8f{0.f, 0.f, 0.f, 0.f, 0.f, 0.f, 0.f, 0.f};
#pragma unroll
    for (int c = 0; c < 32; ++c) {
      const float2 kk = *(const float2*)(kfr + 4 * c + 2 * sel);
      v2f a;  a[0] = kk.x;     a[1] = kk.y;
      v2f bq; bq[0] = qf[c].x; bq[1] = qf[c].y;
      s = __builtin_amdgcn_wmma_f32_16x16x4_f32(false, a, false, bq,
                                                (short)0, s, false, false);
    }

    // ---- mask + online softmax (lane owns head h, tokens r+8*sel) ----
    float sv[8];
    float mt = NEGINF_;
#pragma unroll
    for (int r = 0; r < 8; ++r) {
      const int tok = t0 + 16 * i + r + 8 * sel;
      const float x = (tok < t1) ? s[r] : NEGINF_;
      sv[r] = x;
      mt = fmaxf(mt, x);
    }
    mt = fmaxf(mt, __shfl_xor(mt, 16, 32));
    const float mnew = fmaxf(m_run, mt);
    const float scl = __expf(m_run - mnew);
    float p[8];
    float lt = 0.f;
#pragma unroll
    for (int r = 0; r < 8; ++r) {
      p[r] = __expf(sv[r] - mnew);
      lt += p[r];
    }
    lt += __shfl_xor(lt, 16, 32);
    l_run = l_run * scl + lt;
    m_run = mnew;
#pragma unroll
    for (int cc = 0; cc < 8; ++cc)
#pragma unroll
      for (int j = 0; j < 8; ++j) oacc[cc][j] *= scl;

    // ---- O^T += V^T(128x16) x P^T(16x16heads) ----
    float sxp[8];
#pragma unroll
    for (int r = 0; r < 8; ++r) sxp[r] = __shfl_xor(p[r], 16, 32);
#pragma unroll
    for (int kc4 = 0; kc4 < 4; ++kc4) {
      const int kc = 4 * kc4;
      // P^T rows kc..kc+3 redistributed into B-operand layout
      auto ptlo = [&](int t) { return (t < 8) ? p[t] : sxp[t - 8]; };
      auto pthi = [&](int t) { return (t < 8) ? sxp[t] : p[t - 8]; };
      v2f bp;
      bp[0] = sel ? pthi(kc + 2) : ptlo(kc);
      bp[1] = sel ? pthi(kc + 3) : ptlo(kc + 1);
      const int ra = kc + 2 * sel;
#pragma unroll
      for (int cc = 0; cc < 8; ++cc) {
        v2f a;
#if HAS_ASYNC_LDS
        a[0] = sm.stage[wave][1][ra][16 * cc + h];
        a[1] = sm.stage[wave][1][ra + 1][16 * cc + h];
#else
        a[0] = vbg[(size_t)ra * KVROW_ + 16 * cc + h];
        a[1] = vbg[(size_t)(ra + 1) * KVROW_ + 16 * cc + h];
#endif
        oacc[cc] = __builtin_amdgcn_wmma_f32_16x16x4_f32(
            false, a, false, bp, (short)0, oacc[cc], false, false);
      }
    }
  }

  // ---- splice current token (position N-1): RoPE'd K input + raw V ----
  if (((N - 1) / CHUNK_TOK_) == wave) {
    const float* kp = K + ((size_t)b * HKV_ + kvh) * D_;
    float part = 0.f;
#pragma unroll
    for (int c = 0; c < 32; ++c) {
      const int d0 = 4 * c + 2 * sel;
      const float k0 = kp[d0], k1 = kp[d0 + 1];
      const float r0 = (d0 < 64) ? -kp[d0 + 64] : kp[d0 - 64];
      const float r1 = (d0 < 64) ? -kp[d0 + 65] : kp[d0 - 63];
      const float kr0 = k0 * cb[d0] + r0 * sb[d0];
      const float kr1 = k1 * cb[d0 + 1] + r1 * sb[d0 + 1];
      part += qf[c].x * kr0 + qf[c].y * kr1;
    }
    const float sl = part + __shfl_xor(part, 16, 32);  // full dot, SCALE in qf
    const float mnew = fmaxf(m_run, sl);
    const float scl = __expf(m_run - mnew);
    const float pl = __expf(sl - mnew);
    const float* vp = V + ((size_t)b * HKV_ + kvh) * D_;
#pragma unroll
    for (int cc = 0; cc < 8; ++cc)
#pragma unroll
      for (int j = 0; j < 8; ++j)
        oacc[cc][j] = oacc[cc][j] * scl + pl * vp[16 * cc + 8 * sel + j];
    l_run = l_run * scl + pl;
    m_run = mnew;
  }

  // ---- cross-chunk log-sum-exp combine via LDS (reuses stage area) ----
  __syncthreads();  // all waves done with their stage slices before reuse
#pragma unroll
  for (int cc = 0; cc < 8; ++cc)
#pragma unroll
    for (int j = 0; j < 8; ++j) sm.red[wave][lane][cc * 8 + j] = oacc[cc][j];
  sm.red[wave][lane][64] = m_run;
  sm.red[wave][lane][65] = l_run;
  __syncthreads();

  {
    const int cc = wave;  // each wave finalizes one 16-dim output chunk
    float mstar = NEGINF_;
#pragma unroll
    for (int w = 0; w < NCHUNK_; ++w) mstar = fmaxf(mstar, sm.red[w][lane][64]);
    float wgt[NCHUNK_];
    float Ls = 0.f;
#pragma unroll
    for (int w = 0; w < NCHUNK_; ++w) {
      const float e = __expf(sm.red[w][lane][64] - mstar);
      wgt[w] = e;
      Ls += sm.red[w][lane][65] * e;
    }
    const float inv = 1.0f / Ls;
    if (h < G_) {
      const int hg = h * HKV_ + kvh;
      float* op = out + ((size_t)b * H_ + hg) * D_ + 16 * cc + 8 * sel;
#pragma unroll
      for (int j = 0; j < 8; ++j) {
        float num = 0.f;
#pragma unroll
        for (int w = 0; w < NCHUNK_; ++w)
          num += wgt[w] * sm.red[w][lane][cc * 8 + j];
        op[j] = num * inv;
      }
    }
  }
}

extern "C" void kernel_launch(void* const* d_in, const int* in_sizes, int n_in,
                              void* d_out, int out_size, void* d_ws,
                              size_t ws_size, hipStream_t stream) {
  (void)in_sizes; (void)n_in; (void)d_ws; (void)ws_size; (void)out_size;
  const float* Q = (const float*)d_in[0];
  const float* K = (const float*)d_in[1];
  const float* V = (const float*)d_in[2];
  const float* Kc = (const float*)d_in[3];
  const float* Vc = (const float*)d_in[4];
  const float* cosb = (const float*)d_in[5];
  const float* sinb = (const float*)d_in[6];
  const int* ilen = (const int*)d_in[7];
  const int* fslots = (const int*)d_in[8];
  // d_in[9] = save_slots: not needed for the attention output
  float* out = (float*)d_out;
  dim3 grid(B_ * HKV_), block(256);
  paged_attn_decode<<<grid, block, 0, stream>>>(Q, K, V, Kc, Vc, cosb, sinb,
                                                ilen, fslots, out);
}